// CausalWanSelfAttention_1357209666121
// MI455X (gfx1250) — compile-verified
//
#include <hip/hip_runtime.h>

// ---------------------------------------------------------------------------
// Problem constants (from the reference):
//   x      : [1, 2048, 1024]  fp32
//   w_qkv  : [192, 1024]      fp32   (torch Linear layout: [out, in])
//   w_out  : [1024, 64]       fp32
//   y      : [1, 2048, 1024]  fp32
// per-head dim is 1 (64 heads), scale = 1/sqrt(64) = 0.125
// ---------------------------------------------------------------------------
#define SEQ   2048
#define DIM   1024
#define NHEAD 64
#define FQKV  (3 * NHEAD)   // 192

typedef float v2f __attribute__((ext_vector_type(2)));
typedef float v8f __attribute__((ext_vector_type(8)));

// ---------------------------------------------------------------------------
// C[M x N] = A[M x K] * W[N x K]^T   (all fp32, row-major)
// One wave32 per 16x16 C tile; K-loop steps of 4 using V_WMMA_F32_16X16X4_F32.
//
// VGPR layouts per CDNA5 ISA 7.12.2:
//   A (32-bit, 16x4):  lanes 0-15 -> M=lane, VGPR0=K0,VGPR1=K1
//                      lanes 16-31 -> M=lane-16, VGPR0=K2,VGPR1=K3
//   B (32-bit, 4x16):  symmetric (N=lane&15, K = (lane>>4)*2 + {0,1})
//   C/D (32-bit,16x16): VGPR r: lanes 0-15 -> M=r, lanes 16-31 -> M=8+r; N=lane&15
// ---------------------------------------------------------------------------
__global__ __launch_bounds__(32)
void gemm_f32_wmma(const float* __restrict__ A,
                   const float* __restrict__ W,
                   float* __restrict__ C,
                   int N, int K) {
  const int lane = threadIdx.x;     // 0..31, wave32
  const int half = lane >> 4;       // 0 or 1
  const int sub  = lane & 15;       // A row within tile / B col within tile
  const int m0 = blockIdx.y << 4;
  const int n0 = blockIdx.x << 4;

  const float* __restrict__ arow = A + (size_t)(m0 + sub) * K;
  const float* __restrict__ wrow = W + (size_t)(n0 + sub) * K;

  v8f acc = {};
  for (int k0 = 0; k0 < K; k0 += 4) {
    const int ka = k0 + half * 2;
    v2f a, b;
    a.x = arow[ka];
    a.y = arow[ka + 1];
    b.x = wrow[ka];
    b.y = wrow[ka + 1];
    // (neg_a, A, neg_b, B, c_mod, C, reuse_a, reuse_b)
    acc = __builtin_amdgcn_wmma_f32_16x16x4_f32(false, a, false, b,
                                                (short)0, acc, false, false);
  }

#pragma unroll
  for (int r = 0; r < 8; ++r) {
    C[(size_t)(m0 + r + half * 8) * N + (n0 + sub)] = acc[r];
  }
}

// ---------------------------------------------------------------------------
// Per-head causal attention with head-dim 1 (the reference's quirk):
//   oh[i,h] = sum_{j<=i} softmax_j(q[i,h] * k[j,h] * 0.125) * v[j,h]
// One block = one head x 256 queries. k/v columns of this head staged in LDS,
// one lane per query runs the online-softmax scan over j <= i.
// ---------------------------------------------------------------------------
__global__ __launch_bounds__(256)
void attn_headdim1_causal(const float* __restrict__ qkv,  // [SEQ, 192]
                          float* __restrict__ oh) {       // [SEQ, 64]
  __shared__ float k_s[SEQ];
  __shared__ float v_s[SEQ];
  const int h   = blockIdx.y;
  const int tid = threadIdx.x;

  for (int t = tid; t < SEQ; t += 256) {
    const size_t base = (size_t)t * FQKV;
    k_s[t] = qkv[base + NHEAD + h];
    v_s[t] = qkv[base + 2 * NHEAD + h];
  }
  __syncthreads();

  const int i = blockIdx.x * 256 + tid;
  const float q = qkv[(size_t)i * FQKV + h] * 0.125f;

  float m = -3.402823466e38f;
  float l = 0.0f;
  float acc = 0.0f;
  for (int j = 0; j <= i; ++j) {
    const float s    = q * k_s[j];
    const float mn   = fmaxf(m, s);
    const float corr = __expf(m - mn);
    const float p    = __expf(s - mn);
    l   = l * corr + p;
    acc = acc * corr + p * v_s[j];
    m   = mn;
  }
  oh[(size_t)i * NHEAD + h] = acc / l;
}

// ---------------------------------------------------------------------------
// Launch: qkv GEMM -> attention -> output GEMM. Scratch in d_ws:
//   qkv : SEQ*192 floats (1.5 MB)
//   oh  : SEQ*64  floats (0.5 MB)
// ---------------------------------------------------------------------------
extern "C" void kernel_launch(void* const* d_in, const int* in_sizes, int n_in,
                              void* d_out, int out_size, void* d_ws, size_t ws_size,
                              hipStream_t stream) {
  const float* x     = (const float*)d_in[0];   // [1, 2048, 1024]
  const float* w_qkv = (const float*)d_in[1];   // [192, 1024]
  const float* w_out = (const float*)d_in[2];   // [1024, 64]
  float* y = (float*)d_out;                     // [1, 2048, 1024]

  float* qkv = (float*)d_ws;                    // [2048, 192]
  float* oh  = qkv + (size_t)SEQ * FQKV;        // [2048, 64]

  // qkv = x * w_qkv^T    (M=2048, N=192, K=1024)
  gemm_f32_wmma<<<dim3(FQKV / 16, SEQ / 16), 32, 0, stream>>>(
      x, w_qkv, qkv, FQKV, DIM);

  // per-head causal softmax attention (head dim 1)
  attn_headdim1_causal<<<dim3(SEQ / 256, NHEAD), 256, 0, stream>>>(qkv, oh);

  // y = oh * w_out^T     (M=2048, N=1024, K=64)
  gemm_f32_wmma<<<dim3(DIM / 16, SEQ / 16), 32, 0, stream>>>(
      oh, w_out, y, DIM, NHEAD);
}